// FFNN_37194416783922
// MI455X (gfx1250) — compile-verified
//
#include <hip/hip_runtime.h>
#include <hip/hip_bf16.h>

// ---------------------------------------------------------------------------
// FFNN inference, algebraically collapsed (exact: network is linear before the
// final sigmoid):
//   w_eff[k] = sum_j W1[j,k] * Wf[1,j]        (k < 300)
//   b_eff    = dot(b1, Wf[1]) + bf[1]
//   t_eff[v] = dot(table[v], w_eff)           (v < 100000)   <- WMMA f32 16x16x4
//   out[b]   = sigmoid( mean_s t_eff[x[b,s]] + b_eff )
// Traffic ~125 MB total vs ~1.2 GB for the naive gather+GEMM path.
// ---------------------------------------------------------------------------

#define VOCAB  100000
#define EMBED  300
#define SEQ    60
#define HIDDEN 512
#define BATCH  16384

typedef float v2f __attribute__((ext_vector_type(2)));
typedef float v8f __attribute__((ext_vector_type(8)));

// Workspace layout (floats): [0..299] = w_eff, [300] = b_eff,
// byte offset 4096 onward: t_eff[VOCAB]  (~404 KB total).
#define WS_TEFF_FLOAT_OFF 1024

// ---------------------------------------------------------------------------
// Kernel 1: fold the fc layer into the input layer.  One block, 512 threads.
// ---------------------------------------------------------------------------
__global__ void k_weff(const float* __restrict__ W1,   // [HIDDEN, EMBED] row-major
                       const float* __restrict__ b1,   // [HIDDEN]
                       const float* __restrict__ Wf,   // [2, HIDDEN] row-major
                       const float* __restrict__ bf,   // [2]
                       float* __restrict__ ws) {
  __shared__ float red[HIDDEN];
  const int t = threadIdx.x;

  // b_eff = dot(b1, Wf[1]) + bf[1]   (tree reduction in LDS)
  red[t] = b1[t] * Wf[HIDDEN + t];
  __syncthreads();
  for (int off = HIDDEN / 2; off > 0; off >>= 1) {
    if (t < off) red[t] += red[t + off];
    __syncthreads();
  }
  if (t == 0) ws[EMBED] = red[0] + bf[1];

  // w_eff[k] = sum_j W1[j*EMBED + k] * Wf[HIDDEN + j]
  // For fixed j, consecutive threads read consecutive addresses (coalesced);
  // Wf[HIDDEN + j] is wave-uniform (scalar load).
  if (t < EMBED) {
    float acc = 0.f;
    for (int j = 0; j < HIDDEN; ++j) {
      acc += W1[(size_t)j * EMBED + t] * Wf[HIDDEN + j];
    }
    ws[t] = acc;
  }
}

// ---------------------------------------------------------------------------
// Kernel 2: t_eff[v] = dot(table[v], w_eff) via V_WMMA_F32_16X16X4_F32.
// One wave handles a 16-row vocab tile; B-matrix = w_eff replicated across all
// 16 columns, so D is column-constant and lanes 0/16 hold M=0..7 / M=8..15.
// A-layout (32-bit A 16x4): lanes 0-15 -> K={0,1}, lanes 16-31 -> K={2,3},
// i.e. each lane loads an 8-byte-aligned float2 from its row.
// VOCAB % 16 == 0 and EMBED % 4 == 0, so no tail handling.
// ---------------------------------------------------------------------------
__global__ void k_proj(const float* __restrict__ table,  // [VOCAB, EMBED]
                       const float* __restrict__ ws,
                       float* __restrict__ t_eff) {
  __shared__ __align__(16) float lw[EMBED + 4];
  for (int i = threadIdx.x; i < EMBED; i += blockDim.x) lw[i] = ws[i];
  __syncthreads();

  const int wave = threadIdx.x >> 5;
  const int lane = threadIdx.x & 31;
  const int tile = blockIdx.x * (blockDim.x >> 5) + wave;
  if (tile >= VOCAB / 16) return;   // wave-uniform: EXEC stays all-ones

  const int half = lane >> 4;       // 0: K pair {0,1}, 1: K pair {2,3}
  const int m    = lane & 15;       // row within tile
  const size_t row = (size_t)tile * 16 + m;
  const float* __restrict__ ap = table + row * EMBED + 2 * half;
  const float* __restrict__ bp = lw + 2 * half;

  v8f c = {};
#pragma unroll 5
  for (int i = 0; i < EMBED / 4; ++i) {
    v2f a = *(const v2f*)(ap + i * 4);   // 8B aligned: row*300 + 4i + 2*half even
    v2f b = *(const v2f*)(bp + i * 4);   // LDS, 8B aligned
    c = __builtin_amdgcn_wmma_f32_16x16x4_f32(
        /*neg_a=*/false, a, /*neg_b=*/false, b,
        /*c_mod=*/(short)0, c, /*reuse_a=*/false, /*reuse_b=*/false);
  }

  // C/D layout: VGPR r, lanes 0-15 -> M=r (N=lane), lanes 16-31 -> M=8+r.
  // Columns are identical, so lane 0 / lane 16 carry the 16 dot products.
  if ((lane & 15) == 0) {
    const int mb = tile * 16 + half * 8;
#pragma unroll
    for (int r = 0; r < 8; ++r) t_eff[mb + r] = c[r];
  }
}

// ---------------------------------------------------------------------------
// Kernel 3: out[b] = sigmoid( mean_s t_eff[x[b,s]] + b_eff ).
// t_eff is 400 KB -> L2/WGP$-resident; index reads are sequential per thread.
// ---------------------------------------------------------------------------
__global__ void k_out(const int* __restrict__ x,      // [BATCH, SEQ]
                      const float* __restrict__ ws,
                      float* __restrict__ out) {
  const int b = blockIdx.x * blockDim.x + threadIdx.x;
  if (b >= BATCH) return;
  const float* __restrict__ t_eff = ws + WS_TEFF_FLOAT_OFF;
  const float  b_eff = ws[EMBED];

  const int* __restrict__ xr = x + (size_t)b * SEQ;
  float s = 0.f;
#pragma unroll 4
  for (int i = 0; i < SEQ; ++i) s += t_eff[xr[i]];

  const float z = s * (1.0f / (float)SEQ) + b_eff;
  out[b] = 1.0f / (1.0f + __expf(-z));
}

// ---------------------------------------------------------------------------
extern "C" void kernel_launch(void* const* d_in, const int* in_sizes, int n_in,
                              void* d_out, int out_size, void* d_ws, size_t ws_size,
                              hipStream_t stream) {
  // setup_inputs order: x, table, W1, b1, Wf, bf
  const int*   x     = (const int*)  d_in[0];
  const float* table = (const float*)d_in[1];
  const float* W1    = (const float*)d_in[2];
  const float* b1    = (const float*)d_in[3];
  const float* Wf    = (const float*)d_in[4];
  const float* bf    = (const float*)d_in[5];
  float* out = (float*)d_out;
  float* ws  = (float*)d_ws;            // needs ~404 KB scratch
  float* t_eff = ws + WS_TEFF_FLOAT_OFF;

  // 1) fold fc into input layer
  k_weff<<<1, HIDDEN, 0, stream>>>(W1, b1, Wf, bf, ws);

  // 2) project vocab table: 6250 16-row tiles, 8 waves (tiles) per block
  const int tiles  = VOCAB / 16;                 // 6250
  const int wblk   = 8;                          // waves per block
  const int blocks = (tiles + wblk - 1) / wblk;  // 782
  k_proj<<<blocks, wblk * 32, 0, stream>>>(table, ws, t_eff);

  // 3) gather + mean + sigmoid
  k_out<<<(BATCH + 255) / 256, 256, 0, stream>>>(x, ws, out);
}